// AugLUT_3710851743827
// MI455X (gfx1250) — compile-verified
//
#include <hip/hip_runtime.h>
#include <hip/hip_bf16.h>

#define N_BINS 20
#define STRENGTH 1.0f
#define EPS_V 1e-5f
#define TAB_STRIDE 64  // floats reserved per batch in workspace (38 used)

typedef __attribute__((ext_vector_type(4))) float v4f;   // clang vector: OK for NT builtins

// ---------------------------------------------------------------------------
// Kernel 1: build per-batch LUT (normalize + convert to intercept/slope form)
// out = a[i] + b[i] * t  where t = x*19, i = clamp(floor(t),0,18)
// a[i] = y[i] - (y[i+1]-y[i])*i ;  b[i] = y[i+1]-y[i]
// ---------------------------------------------------------------------------
__global__ void AugLUT_prep_kernel(const float* __restrict__ ran_y,
                                   float* __restrict__ tab, int bs) {
    int b = blockIdx.x * blockDim.x + threadIdx.x;
    if (b >= bs) return;

    float y[N_BINS];
    float ymin = INFINITY, ymax = -INFINITY;
#pragma unroll
    for (int j = 0; j < N_BINS; ++j) {
        float lin = (float)j / (float)(N_BINS - 1);
        float v = ran_y[b * N_BINS + j] * STRENGTH + lin * (1.0f - STRENGTH);
        y[j] = v;
        ymin = fminf(ymin, v);
        ymax = fmaxf(ymax, v);
    }
    float inv = 1.0f / (ymax - ymin + EPS_V);
#pragma unroll
    for (int j = 0; j < N_BINS; ++j) y[j] = (y[j] - ymin) * inv;

    float* tb = tab + b * TAB_STRIDE;
#pragma unroll
    for (int i = 0; i < N_BINS - 1; ++i) {
        float slope = y[i + 1] - y[i];
        tb[2 * i]     = y[i] - slope * (float)i;  // intercept in t-space
        tb[2 * i + 1] = slope;
    }
}

// ---------------------------------------------------------------------------
// Kernel 2: streaming LUT apply. HBM-bound: b128 non-temporal load/store,
// per-lane ds_load_b64 LUT gather, global_prefetch of next stride.
// ---------------------------------------------------------------------------
__device__ __forceinline__ float apply1(const float2* __restrict__ lut, float xv) {
    float t = xv * (float)(N_BINS - 1);
    int i = (int)t;
    i = (i < 0) ? 0 : i;
    i = (i > N_BINS - 2) ? (N_BINS - 2) : i;
    float2 ab = lut[i];                 // ds_load_b64
    return fmaf(ab.y, t, ab.x);
}

__global__ __launch_bounds__(256) void AugLUT_apply_kernel(
        const v4f* __restrict__ x, const float* __restrict__ tab,
        v4f* __restrict__ out, int quads_per_batch) {
    __shared__ float2 lut[N_BINS - 1];          // 152 B per workgroup

    const int batch = blockIdx.y;
    const int tid = threadIdx.x;
    if (tid < (N_BINS - 1)) {
        const float2* src = (const float2*)(tab + (size_t)batch * TAB_STRIDE);
        lut[tid] = src[tid];
    }
    __syncthreads();

    const v4f* __restrict__ xb = x   + (size_t)batch * quads_per_batch;
    v4f* __restrict__       ob = out + (size_t)batch * quads_per_batch;

    const int stride = gridDim.x * blockDim.x;
    for (int i = blockIdx.x * blockDim.x + tid; i < quads_per_batch; i += stride) {
        // prefetch next grid-stride iteration's cacheline (speculative; OOB ok)
        __builtin_prefetch(&xb[i + stride], 0, 0);

        v4f v = __builtin_nontemporal_load(&xb[i]);   // global_load_b128 NT
        v4f r;
        r[0] = apply1(lut, v[0]);
        r[1] = apply1(lut, v[1]);
        r[2] = apply1(lut, v[2]);
        r[3] = apply1(lut, v[3]);
        __builtin_nontemporal_store(r, &ob[i]);       // global_store_b128 NT
    }
}

// ---------------------------------------------------------------------------
// Launch
// ---------------------------------------------------------------------------
extern "C" void kernel_launch(void* const* d_in, const int* in_sizes, int n_in,
                              void* d_out, int out_size, void* d_ws, size_t ws_size,
                              hipStream_t stream) {
    const float* x     = (const float*)d_in[0];
    const float* ran_y = (const float*)d_in[1];
    float* out = (float*)d_out;
    float* tab = (float*)d_ws;   // needs bs * TAB_STRIDE * 4 bytes (2 KB for bs=8)

    const int bs = in_sizes[1] / N_BINS;          // 8
    const int per_batch = in_sizes[0] / bs;       // 12,582,912 (divisible by 4)
    const int quads = per_batch / 4;              // 3,145,728

    AugLUT_prep_kernel<<<1, 32, 0, stream>>>(ran_y, tab, bs);

    dim3 grid(1024, bs);                          // 8192 blocks, ~12 quads/thread
    AugLUT_apply_kernel<<<grid, 256, 0, stream>>>(
        (const v4f*)x, tab, (v4f*)out, quads);
}